// SymmetryExpansion_67654324846650
// MI455X (gfx1250) — compile-verified
//
#include <hip/hip_runtime.h>
#include <hip/hip_bf16.h>
#include <stdint.h>

// SymmetryExpansion: out[b,k,l,:] = x[b,(l-k)%L,:]  (k <  L)
//                    out[b,k,l,:] = x[b,(k-l)%L,:]  (k >= L)
// B=32, L=128, F=64, fp32.  Pure data movement: 256 MB of stores,
// staged through LDS with CDNA5 async global<->LDS instructions.
// Roofline: 256 MB / 23.3 TB/s ~= 11 us, store-bandwidth bound.

#define B_DIM 32
#define L_DIM 128
#define F_DIM 64
#define KCHUNK 8                      // k-values per block
#define NKC    ((2 * L_DIM) / KCHUNK) // 32 k-chunks
#define ROW_BYTES (F_DIM * 4)         // 256 B per (k,l) row

__global__ __launch_bounds__(256) void symmetry_expansion_async(
    const float* __restrict__ x, float* __restrict__ out) {
  __shared__ __align__(16) float smem[L_DIM * F_DIM]; // 32 KB

  const uint32_t t     = threadIdx.x;              // 0..255
  const uint32_t b     = blockIdx.x / NKC;         // batch (uniform)
  const uint32_t kbase = (blockIdx.x % NKC) * KCHUNK;

  const float* xb  = x + (size_t)b * (L_DIM * F_DIM);               // SGPR base
  float* outb      = out + (size_t)b * (2 * L_DIM * L_DIM * F_DIM); // SGPR base

  const uint32_t lds_base = (uint32_t)(uintptr_t)(&smem[0]); // LDS byte offset

  // ---- Phase 1: async-stage x[b] (32 KB) into LDS, b128 per lane ----
  // 2048 x 16B chunks, 256 threads -> 8 chunks/thread. RT temporal hint:
  // these lines are re-read by the 31 sibling blocks of this batch via L2.
#pragma unroll
  for (int i = 0; i < 8; ++i) {
    uint32_t off = (t + (uint32_t)i * 256u) * 16u;     // byte offset in slice
    uint32_t lds_addr = lds_base + off;
    asm volatile("global_load_async_to_lds_b128 %0, %1, %2"
                 :
                 : "v"(lds_addr), "v"(off), "s"(xb)
                 : "memory");
  }
  asm volatile("s_wait_asynccnt 0" ::: "memory");
  __syncthreads();

  // ---- Phase 2: stream output rows straight from LDS ----
  // Rows this block owns: KCHUNK * L = 1024 rows of 256 B.
  // 16 lanes cover one row (16 B each); 256 threads -> 16 rows/iter.
  // Output is written once and never re-read (256 MB > 192 MB L2), so use
  // non-temporal stores to avoid flushing the reused input out of L2.
  const uint32_t fo = (t & 15u) * 16u;  // byte offset within a row
#pragma unroll 4
  for (uint32_t r = (t >> 4); r < (uint32_t)(KCHUNK * L_DIM); r += 16u) {
    uint32_t kk = r >> 7;          // 0..KCHUNK-1
    uint32_t l  = r & (L_DIM - 1); // 0..127
    uint32_t k  = kbase + kk;      // 0..255
    uint32_t idx = (k < L_DIM) ? ((l - k) & (L_DIM - 1))
                               : ((k - l) & (L_DIM - 1));
    uint32_t lds_src = lds_base + idx * ROW_BYTES + fo;
    uint32_t gdst    = (k * L_DIM + l) * ROW_BYTES + fo; // < 8.4 MB, fits i32
    asm volatile("global_store_async_from_lds_b128 %0, %1, %2 th:TH_STORE_NT"
                 :
                 : "v"(gdst), "v"(lds_src), "s"(outb)
                 : "memory");
  }
  // S_ENDPGM performs an implicit wait-idle; make the drain explicit anyway.
  asm volatile("s_wait_asynccnt 0" ::: "memory");
}

extern "C" void kernel_launch(void* const* d_in, const int* in_sizes, int n_in,
                              void* d_out, int out_size, void* d_ws, size_t ws_size,
                              hipStream_t stream) {
  (void)in_sizes; (void)n_in; (void)d_ws; (void)ws_size; (void)out_size;
  const float* x = (const float*)d_in[0];
  float* out = (float*)d_out;
  dim3 grid(B_DIM * NKC); // 1024 blocks
  dim3 block(256);        // 8 waves, wave32
  symmetry_expansion_async<<<grid, block, 0, stream>>>(x, out);
}